// QCNN_58025008169535
// MI455X (gfx1250) — compile-verified
//
#include <hip/hip_runtime.h>
#include <hip/hip_bf16.h>

typedef float v2f __attribute__((ext_vector_type(2)));
typedef float v8f __attribute__((ext_vector_type(8)));

#define HH 128
#define WW 128
#define CC 3
#define BB 32
#define PH 127
#define PW 127
#define PPB (PH * PW)        // 16129 patches per image
#define TOTAL (BB * PPB)     // 516128 circuits
#define GROUPS 8             // 16-patch WMMA groups per wave

__device__ __forceinline__ void cmul(float ar, float ai, float br, float bi,
                                     float& cr, float& ci) {
    cr = ar * br - ai * bi;
    ci = ar * bi + ai * br;
}

__device__ __forceinline__ v8f wmma4(v2f a, v2f b, v8f c) {
    return __builtin_amdgcn_wmma_f32_16x16x4_f32(false, a, false, b,
                                                 (short)0, c, false, false);
}

// One block = 256 threads = 8 wave32s.  Wave-0 lanes 0..15 build the
// patch-independent 16x16 variational unitary V into LDS once per block;
// then every wave runs GROUPS batches of 16 patches through f32 WMMA.
__global__ __launch_bounds__(256) void qcnn_kernel(const float* __restrict__ x,
                                                   const float* __restrict__ w,
                                                   float* __restrict__ out) {
    __shared__ float sVre[256];
    __shared__ float sVim[256];

    const int tid = threadIdx.x;

    // ---------------- build V = (4 variational layers) as 16x16 complex ----
    if (tid < 16) {
        float vr[16], vi[16];
#pragma unroll
        for (int k = 0; k < 16; ++k) { vr[k] = (k == tid) ? 1.f : 0.f; vi[k] = 0.f; }
#pragma unroll
        for (int h = 0; h < 4; ++h) {
#pragma unroll
            for (int q = 0; q < 4; ++q) {
                const int st = 8 >> q;   // qubit q <-> bit (3-q)
                const float ax = w[(h * 4 + q) * 3 + 0];
                const float ay = w[(h * 4 + q) * 3 + 1];
                const float az = w[(h * 4 + q) * 3 + 2];
                float s, c;
                // Rx(ax)
                __sincosf(0.5f * ax, &s, &c);
#pragma unroll
                for (int k = 0; k < 16; ++k) if (!(k & st)) {
                    float r0 = vr[k], i0 = vi[k], r1 = vr[k + st], i1 = vi[k + st];
                    vr[k]      =  c * r0 + s * i1;
                    vi[k]      =  c * i0 - s * r1;
                    vr[k + st] =  s * i0 + c * r1;
                    vi[k + st] = -s * r0 + c * i1;
                }
                // Ry(ay)
                __sincosf(0.5f * ay, &s, &c);
#pragma unroll
                for (int k = 0; k < 16; ++k) if (!(k & st)) {
                    float r0 = vr[k], i0 = vi[k], r1 = vr[k + st], i1 = vi[k + st];
                    vr[k]      = c * r0 - s * r1;
                    vi[k]      = c * i0 - s * i1;
                    vr[k + st] = s * r0 + c * r1;
                    vi[k + st] = s * i0 + c * i1;
                }
                // Rz(az)
                __sincosf(0.5f * az, &s, &c);
#pragma unroll
                for (int k = 0; k < 16; ++k) if (!(k & st)) {
                    float r0 = vr[k], i0 = vi[k], r1 = vr[k + st], i1 = vi[k + st];
                    vr[k]      = c * r0 + s * i0;
                    vi[k]      = c * i0 - s * r0;
                    vr[k + st] = c * r1 - s * i1;
                    vi[k + st] = c * i1 + s * r1;
                }
            }
            // CZ(0,1), CZ(2,3), CZ(1,2): net sign = (-1)^(q0q1 ^ q2q3 ^ q1q2)
#pragma unroll
            for (int k = 0; k < 16; ++k) {
                int q0 = (k >> 3) & 1, q1 = (k >> 2) & 1, q2 = (k >> 1) & 1, q3 = k & 1;
                if ((q0 & q1) ^ (q2 & q3) ^ (q1 & q2)) { vr[k] = -vr[k]; vi[k] = -vi[k]; }
            }
        }
#pragma unroll
        for (int m = 0; m < 16; ++m) {
            sVre[m * 16 + tid] = vr[m];   // V[m][col]
            sVim[m * 16 + tid] = vi[m];
        }
    }
    __syncthreads();

    // ---------------- per-wave WMMA batches of 16 patches --------------------
    const int lane = tid & 31;
    const int wv   = tid >> 5;
    const int col  = lane & 15;        // N column = patch slot, also A-row m
    const int hi   = lane >> 4;        // lane half: fragment K offset (+2), q2 bit

    // A fragments (V) from LDS, register-resident for the whole group loop:
    // VGPR v holds V[col][4*kk + 2*hi + v]
    v2f aRe[4], aIm[4], aImN[4];
    {
        const float2* sVre2 = (const float2*)sVre;
        const float2* sVim2 = (const float2*)sVim;
#pragma unroll
        for (int kk = 0; kk < 4; ++kk) {
            const int k0 = 4 * kk + 2 * hi;           // even -> 8B-aligned ds_load_b64
            float2 re = sVre2[(col * 16 + k0) >> 1];
            float2 im = sVim2[(col * 16 + k0) >> 1];
            aRe[kk][0]  = re.x;  aRe[kk][1]  = re.y;
            aIm[kk][0]  = im.x;  aIm[kk][1]  = im.y;
            aImN[kk][0] = -im.x; aImN[kk][1] = -im.y; // f32 WMMA has no A-neg modifier
        }
    }

    const int waveGlobal = (int)blockIdx.x * 8 + wv;

#pragma unroll 1
    for (int g = 0; g < GROUPS; ++g) {
        const int gpatch = (waveGlobal * GROUPS + g) * 16 + col;
        const bool valid = gpatch < TOTAL;
        const int p  = valid ? gpatch : TOTAL - 1;    // clamp: keep EXEC full for WMMA
        const int b  = p / PPB;
        const int rr = p - b * PPB;
        const int py = rr / PW;
        const int px = rr - py * PW;

        // Lane pair (col, col+16) shares patch p.  hi=0 lane encodes qubits 0,1
        // (pixel row py), hi=1 lane encodes qubits 2,3 (pixel row py+1); the
        // leading Rz of the encoding is a global phase and is dropped, so only
        // channels 1 (Ry angle) and 2 (Rz angle) are needed.
        float mur[2][2], mui[2][2];                   // my two qubits: [t][state]
#pragma unroll
        for (int t = 0; t < 2; ++t) {
            const int iy = py + hi;
            const int ix = px + t;
            const float* pix = x + (((size_t)b * HH + iy) * WW + ix) * CC;
            const float gch = pix[1];
            const float bch = pix[2];
            const float sb = sinpif(0.5f * gch), cb = cospif(0.5f * gch);
            const float sc = sinpif(0.5f * bch), cc = cospif(0.5f * bch);
            mur[t][0] = cb * cc;  mui[t][0] = -cb * sc;   // e^{-ic/2} cos(b/2)
            mur[t][1] = sb * cc;  mui[t][1] =  sb * sc;   // e^{+ic/2} sin(b/2)
        }
        // exchange with partner lane; assemble qubits 0,1 and 2,3
        float u01r[2][2], u01i[2][2], u23r[2][2], u23i[2][2];
#pragma unroll
        for (int t = 0; t < 2; ++t)
#pragma unroll
            for (int s = 0; s < 2; ++s) {
                const float tr = __shfl_xor(mur[t][s], 16, 32);
                const float ti = __shfl_xor(mui[t][s], 16, 32);
                u01r[t][s] = hi ? tr : mur[t][s];
                u01i[t][s] = hi ? ti : mui[t][s];
                u23r[t][s] = hi ? mur[t][s] : tr;
                u23i[t][s] = hi ? mui[t][s] : ti;
            }

        // B fragments: global k = 4*kk + 2*hi + v  =>  q0q1 = kk, q2 = hi, q3 = v
        const float f2r = u23r[0][hi], f2i = u23i[0][hi];
        v2f bRe[4], bIm[4];
#pragma unroll
        for (int kk = 0; kk < 4; ++kk) {
            const int q0 = kk >> 1, q1 = kk & 1;
            float p01r, p01i, p2r, p2i, a0r, a0i, a1r, a1i;
            cmul(u01r[0][q0], u01i[0][q0], u01r[1][q1], u01i[1][q1], p01r, p01i);
            cmul(p01r, p01i, f2r, f2i, p2r, p2i);
            cmul(p2r, p2i, u23r[1][0], u23i[1][0], a0r, a0i);
            cmul(p2r, p2i, u23r[1][1], u23i[1][1], a1r, a1i);
            bRe[kk][0] = a0r; bRe[kk][1] = a1r;
            bIm[kk][0] = a0i; bIm[kk][1] = a1i;
        }

        // psi = V * psi_enc: 4 independent 4-deep WMMA chains (K 0..7 / 8..15)
        v8f accRe0 = {0, 0, 0, 0, 0, 0, 0, 0};
        v8f accRe1 = {0, 0, 0, 0, 0, 0, 0, 0};
        v8f accIm0 = {0, 0, 0, 0, 0, 0, 0, 0};
        v8f accIm1 = {0, 0, 0, 0, 0, 0, 0, 0};
#pragma unroll
        for (int kk = 0; kk < 2; ++kk) {
            accRe0 = wmma4(aRe[kk],     bRe[kk],     accRe0);
            accRe1 = wmma4(aRe[kk + 2], bRe[kk + 2], accRe1);
            accIm0 = wmma4(aRe[kk],     bIm[kk],     accIm0);
            accIm1 = wmma4(aRe[kk + 2], bIm[kk + 2], accIm1);
        }
#pragma unroll
        for (int kk = 0; kk < 2; ++kk) {
            accRe0 = wmma4(aImN[kk],     bIm[kk],     accRe0);
            accRe1 = wmma4(aImN[kk + 2], bIm[kk + 2], accRe1);
            accIm0 = wmma4(aIm[kk],      bRe[kk],     accIm0);
            accIm1 = wmma4(aIm[kk + 2],  bRe[kk + 2], accIm1);
        }

        // probabilities; lane holds states m + 8*hi for m = 0..7
        float pr[8];
#pragma unroll
        for (int m = 0; m < 8; ++m) {
            const float re = accRe0[m] + accRe1[m];
            const float im = accIm0[m] + accIm1[m];
            pr[m] = re * re + im * im;
        }

        // signed partial sums for <Z_i>; qubit0 bit == lane-half bit (hi)
        float S0 = 0.f, S1 = 0.f, S2 = 0.f, S3 = 0.f;
#pragma unroll
        for (int m = 0; m < 8; ++m) {
            S0 += pr[m];
            S1 += ((m >> 2) & 1) ? -pr[m] : pr[m];
            S2 += ((m >> 1) & 1) ? -pr[m] : pr[m];
            S3 += (m & 1)        ? -pr[m] : pr[m];
        }
        const float T0 = __shfl_xor(S0, 16, 32);
        const float T1 = __shfl_xor(S1, 16, 32);
        const float T2 = __shfl_xor(S2, 16, 32);
        const float T3 = __shfl_xor(S3, 16, 32);
        const float z0 = S0 - T0;      // <Z_0>: + for qubit0=0 half, - for =1 half
        const float z1 = S1 + T1;
        const float z2 = S2 + T2;
        const float z3 = S3 + T3;

        if (hi == 0 && valid) {
            float4 o = make_float4(z0, z1, z2, z3);
            *(float4*)(out + (size_t)gpatch * 4) = o;   // out[b, py*127+px, 0..3]
        }
    }
}

extern "C" void kernel_launch(void* const* d_in, const int* in_sizes, int n_in,
                              void* d_out, int out_size, void* d_ws, size_t ws_size,
                              hipStream_t stream) {
    (void)in_sizes; (void)n_in; (void)out_size; (void)d_ws; (void)ws_size;
    const float* x = (const float*)d_in[0];   // [32,128,128,3] f32
    const float* w = (const float*)d_in[1];   // [4,4,3] f32
    float* out = (float*)d_out;               // [32,16129,4] f32

    const int patchesPerBlock = 8 * 16 * GROUPS;   // 8 waves x 16 patches x GROUPS
    const int blocks = (TOTAL + patchesPerBlock - 1) / patchesPerBlock;
    qcnn_kernel<<<blocks, 256, 0, stream>>>(x, w, out);
}